// ClassAwareTripletLoss_5952824672892
// MI455X (gfx1250) — compile-verified
//
#include <hip/hip_runtime.h>
#include <hip/hip_bf16.h>

typedef __attribute__((ext_vector_type(2))) float v2f;
typedef __attribute__((ext_vector_type(8))) float v8f;

#define N_ROWS 4096
#define DIM    128
#define MARGIN1 0.3f
#define MARGIN2 0.1f
#define EPS_D2  1e-12f

// ---------------------------------------------------------------------------
// Kernel 1: per-row squared norms  sq[i] = sum_k X[i][k]^2
// ---------------------------------------------------------------------------
__global__ void catl_rowsq(const float* __restrict__ X, float* __restrict__ sq) {
  int i = blockIdx.x * blockDim.x + threadIdx.x;
  if (i >= N_ROWS) return;
  const float4* p = (const float4*)(X + (size_t)i * DIM);
  float s = 0.f;
#pragma unroll
  for (int k = 0; k < DIM / 4; ++k) {
    float4 v = p[k];
    s += v.x * v.x + v.y * v.y + v.z * v.z + v.w * v.w;
  }
  sq[i] = s;
}

// ---------------------------------------------------------------------------
// Kernel 2: fused Gram (fp32 WMMA) + distance + masked row reductions.
// One workgroup (4 waves) per 16-anchor block; waves stride over column tiles.
//
// WMMA fp32 16x16x4 fragment layout (wave32):
//   A (16x4):  lanes 0-15 -> M = lane,  VGPR0/1 = K0/K1
//              lanes 16-31 -> M = lane-16, VGPR0/1 = K2/K3
//   B (4x16):  lanes 0-15 -> N = lane,  VGPR0/1 = K0/K1 (symmetric, K via lane[4])
//   C/D (16x16): VGPR r -> M = r + 8*(lane>>4), N = lane & 15
// ---------------------------------------------------------------------------
__launch_bounds__(128)
__global__ void catl_tiles(const float* __restrict__ X,
                           const int*   __restrict__ tgt,
                           const int*   __restrict__ mac,
                           const float* __restrict__ sq,
                           float*       __restrict__ out,
                           float*       __restrict__ partials) {
  const int bi   = blockIdx.x;            // anchor block: rows [bi*16, bi*16+16)
  const int lane = threadIdx.x & 31;
  const int wave = threadIdx.x >> 5;      // 0..3
  const int n    = lane & 15;
  const int hi   = lane >> 4;             // 0 or 1
  const int i0   = bi * 16;

  // ---- preload A fragments for this anchor block (reused for all j tiles) ----
  const int rowA = i0 + n;                // M = lane & 15
  const int kh   = hi * 2;                // K offset within each group of 4
  v2f afrag[32];
#pragma unroll
  for (int kk = 0; kk < 32; ++kk)
    afrag[kk] = *(const v2f*)(X + (size_t)rowA * DIM + 4 * kk + kh);

  // ---- per-lane metadata for the 8 anchor rows this lane accumulates ----
  float sqi[8]; int ti[8]; int mi[8];
#pragma unroll
  for (int r = 0; r < 8; ++r) {
    int i = i0 + r + 8 * hi;
    sqi[r] = sq[i]; ti[r] = tgt[i]; mi[r] = mac[i];
  }

  const float INF = __builtin_inff();
  float ap[8], anyn[8], intra[8], inter[8];
#pragma unroll
  for (int r = 0; r < 8; ++r) { ap[r] = -INF; anyn[r] = INF; intra[r] = INF; inter[r] = INF; }

  // ---- sweep column tiles, wave-strided ----
  for (int jb = wave; jb < N_ROWS / 16; jb += 4) {
    const int j0   = jb * 16;
    const int rowB = j0 + n;              // N = lane & 15
    const float* bp = X + (size_t)rowB * DIM + kh;

    v8f acc = {};
#pragma unroll
    for (int kk = 0; kk < 32; ++kk) {
      v2f b = *(const v2f*)(bp + 4 * kk);
      // D = A x B + C, fp32 exact (matches reference fp32 matmul)
      acc = __builtin_amdgcn_wmma_f32_16x16x4_f32(false, afrag[kk], false, b,
                                                  (short)0, acc, false, false);
    }

    const int   j   = j0 + n;             // this lane's column
    const float sqj = sq[j];
    const int   tj  = tgt[j];
    const int   mj  = mac[j];
#pragma unroll
    for (int r = 0; r < 8; ++r) {
      float d2 = sqi[r] + sqj - 2.0f * acc[r];
      float d  = sqrtf(fmaxf(d2, EPS_D2));
      bool same_t = (ti[r] == tj);
      bool same_m = (mi[r] == mj);
      ap[r]    = same_t              ? fmaxf(ap[r], d)    : ap[r];
      anyn[r]  = (!same_t)           ? fminf(anyn[r], d)  : anyn[r];
      intra[r] = (!same_t && same_m) ? fminf(intra[r], d) : intra[r];
      inter[r] = (!same_m)           ? fminf(inter[r], d) : inter[r];
    }
  }

  // ---- butterfly reduce across the 16 lanes of each half-wave ----
#pragma unroll
  for (int r = 0; r < 8; ++r) {
#pragma unroll
    for (int m = 1; m <= 8; m <<= 1) {
      ap[r]    = fmaxf(ap[r],    __shfl_xor(ap[r],    m, 32));
      anyn[r]  = fminf(anyn[r],  __shfl_xor(anyn[r],  m, 32));
      intra[r] = fminf(intra[r], __shfl_xor(intra[r], m, 32));
      inter[r] = fminf(inter[r], __shfl_xor(inter[r], m, 32));
    }
  }

  // ---- combine the 4 waves through LDS ----
  __shared__ float s_ap[4][16], s_any[4][16], s_in[4][16], s_ie[4][16];
  __shared__ float s_loss[16];
  if (n == 0) {
#pragma unroll
    for (int r = 0; r < 8; ++r) {
      int m = r + 8 * hi;
      s_ap[wave][m]  = ap[r];
      s_any[wave][m] = anyn[r];
      s_in[wave][m]  = intra[r];
      s_ie[wave][m]  = inter[r];
    }
  }
  __syncthreads();

  if (threadIdx.x < 16) {
    int m = threadIdx.x;
    float fap = -INF, fany = INF, fin = INF, fie = INF;
#pragma unroll
    for (int w = 0; w < 4; ++w) {
      fap  = fmaxf(fap,  s_ap[w][m]);
      fany = fminf(fany, s_any[w][m]);
      fin  = fminf(fin,  s_in[w][m]);
      fie  = fminf(fie,  s_ie[w][m]);
    }
    float an  = (fin < INF) ? fin : fany;   // intra-macro min, fallback any-neg
    float ani = (fie < INF) ? fie : fany;   // inter-macro min, fallback any-neg
    int i = i0 + m;
    out[1 + i]          = fap;              // dist_ap
    out[1 + N_ROWS + i] = an;               // dist_an
    s_loss[m] = fmaxf(0.f, MARGIN1 - (an  - fap)) +
                fmaxf(0.f, MARGIN2 - (ani - fap));
  }
  __syncthreads();

  if (threadIdx.x == 0) {
    float t = 0.f;
#pragma unroll
    for (int m = 0; m < 16; ++m) t += s_loss[m];
    partials[bi] = t;
  }
}

// ---------------------------------------------------------------------------
// Kernel 3: deterministic tree-reduce of 256 block partials -> loss
// ---------------------------------------------------------------------------
__global__ void catl_finish(const float* __restrict__ partials, float* __restrict__ out) {
  __shared__ float s[256];
  int t = threadIdx.x;
  s[t] = partials[t];
  __syncthreads();
  for (int w = 128; w > 0; w >>= 1) {
    if (t < w) s[t] += s[t + w];
    __syncthreads();
  }
  if (t == 0) out[0] = s[0] * (1.0f / (float)N_ROWS);
}

// ---------------------------------------------------------------------------
extern "C" void kernel_launch(void* const* d_in, const int* in_sizes, int n_in,
                              void* d_out, int out_size, void* d_ws, size_t ws_size,
                              hipStream_t stream) {
  const float* X   = (const float*)d_in[0];
  const int*   tgt = (const int*)d_in[1];
  const int*   mac = (const int*)d_in[2];
  float* out = (float*)d_out;

  float* sq       = (float*)d_ws;       // N_ROWS floats
  float* partials = sq + N_ROWS;        // 256 floats

  catl_rowsq <<<N_ROWS / 256, 256, 0, stream>>>(X, sq);
  catl_tiles <<<N_ROWS / 16,  128, 0, stream>>>(X, tgt, mac, sq, out, partials);
  catl_finish<<<1,            256, 0, stream>>>(partials, out);
}